// PostProcessingBlock_74990128988256
// MI455X (gfx1250) — compile-verified
//
#include <hip/hip_runtime.h>

// ---------------------------------------------------------------------------
// DAG-prox iteration (B=512, N=64, 50 iters) for MI455X / gfx1250.
//
// Compute-bound: ~230 GFLOP fp32 vs ~16 MB total HBM traffic (sub-µs at
// 23.3 TB/s). Strategy: 1 workgroup per batch matrix; the whole 50-iteration
// chain stays in LDS (3 rotating 64x64 fp32 buffers, ~50 KB) and all matmuls
// use V_WMMA_F32_16X16X4_F32 (native fp32 WMMA, matches reference precision).
// ---------------------------------------------------------------------------

typedef __attribute__((ext_vector_type(2))) float v2f;
typedef __attribute__((ext_vector_type(8))) float v8f;

#define NDIM      64
#define STRIDE    66            // padded LDS row stride (floats): even (8B-aligned
                                // v2f A-loads) and != 64 (breaks bank-stride conflicts)
#define MATF      (NDIM * STRIDE)
#define NTHREADS  256
#define NWAVES    8
#define NELEM     (NDIM * NDIM) // 4096
#define PER_THR   (NELEM / NTHREADS) // 16 x-elements per thread, held in registers
#define NUM_ITERS 50
#define STEP_PRI  0.01f
#define STEP_DUAL 0.01f
#define REG_SP    0.002f

// 64x64 fp32 matmul D = A @ B entirely in LDS, via v_wmma_f32_16x16x4_f32.
// 16 output tiles of 16x16; wave w owns tiles {w, w+8}; 16 K-steps of K=4.
__device__ __forceinline__ void matmul64(float* __restrict__ D,
                                         const float* __restrict__ A,
                                         const float* __restrict__ B,
                                         int lane, int wave) {
    __syncthreads();                 // inputs fully written by all waves
    const int half = lane >> 4;      // 0: lanes 0-15, 1: lanes 16-31
    const int l16  = lane & 15;
#pragma unroll
    for (int ti = 0; ti < 2; ++ti) {
        const int t  = wave + ti * NWAVES;   // tile id 0..15
        const int mr = (t >> 2) * 16;        // tile row base
        const int nc = (t & 3) * 16;         // tile col base
        v8f acc = {};
        const int arow = (mr + l16) * STRIDE;
        const int bcol = nc + l16;
#pragma unroll
        for (int kk = 0; kk < 16; ++kk) {
            const int k0 = kk * 4 + 2 * half;          // even -> 8B aligned
            // A frag (16x4 f32): lane row = mr+l16, holds K = k0, k0+1
            const v2f a = *(const v2f*)&A[arow + k0];
            // B frag (4x16 f32): lane col = nc+l16, holds K = k0, k0+1
            v2f b;
            b.x = B[k0 * STRIDE + bcol];
            b.y = B[(k0 + 1) * STRIDE + bcol];
            acc = __builtin_amdgcn_wmma_f32_16x16x4_f32(
                false, a, false, b, (short)0, acc, false, false);
        }
        // C/D layout: VGPR r -> row mr + r (lanes 0-15) / mr + r + 8 (16-31)
#pragma unroll
        for (int r = 0; r < 8; ++r) {
            D[(mr + r + 8 * half) * STRIDE + nc + l16] = acc[r];
        }
    }
    __syncthreads();                 // D fully written before next consumer
}

__global__ __launch_bounds__(NTHREADS)
void dag_prox_kernel(const float* __restrict__ adj, float* __restrict__ out) {
    __shared__ float b0[MATF];
    __shared__ float b1[MATF];
    __shared__ float b2[MATF];
    __shared__ float s_alpha;

    const int tid  = threadIdx.x;
    const int lane = tid & 31;
    const int wave = tid >> 5;
    const float* adjb = adj + (size_t)blockIdx.x * NELEM;
    float*       outb = out + (size_t)blockIdx.x * NELEM;

    // x lives in registers: element e = k*256 + tid  ->  r = 4k + tid/64,
    // c = tid & 63 (fixed column per thread).
    const int c0 = tid & 63;
    const int r0 = tid >> 6;
    float xreg[PER_THR];
#pragma unroll
    for (int k = 0; k < PER_THR; ++k) xreg[k] = adjb[k * NTHREADS + tid];
    if (tid == 0) s_alpha = 0.0f;
    __syncthreads();

    for (int it = 0; it < NUM_ITERS; ++it) {
        // ---- m = I + (x*x)/N into b0 (elementwise, owned indices) ----
#pragma unroll
        for (int k = 0; k < PER_THR; ++k) {
            const int r = 4 * k + r0;
            const float v = xreg[k];
            b0[r * STRIDE + c0] = v * v * (1.0f / NDIM) + (r == c0 ? 1.0f : 0.0f);
        }
        // ---- P = m^63, copy-free 3-buffer squaring schedule ----
        matmul64(b1, b0, b0, lane, wave);  // m^2
        matmul64(b2, b0, b1, lane, wave);  // R = m^3      (m dead)
        matmul64(b0, b1, b1, lane, wave);  // m^4
        matmul64(b1, b2, b0, lane, wave);  // R = m^7      (m^2 dead)
        matmul64(b2, b0, b0, lane, wave);  // m^8
        matmul64(b0, b1, b2, lane, wave);  // R = m^15
        matmul64(b1, b2, b2, lane, wave);  // m^16
        matmul64(b2, b0, b1, lane, wave);  // R = m^31
        matmul64(b0, b1, b1, lane, wave);  // m^32
        matmul64(b1, b2, b0, lane, wave);  // P = m^63 -> b1

        // ---- primal prox-gradient step (elementwise, reads P^T) ----
        const float alpha = s_alpha;       // last write was before a barrier
#pragma unroll
        for (int k = 0; k < PER_THR; ++k) {
            const int r = 4 * k + r0;
            const float a  = adjb[k * NTHREADS + tid];
            const float s  = (a > 0.5f) ? a : 0.0f;        // threshold(adj)
            const float xv = xreg[k];
            const float pt = b1[c0 * STRIDE + r];          // P^T[r][c] = P[c][r]
            const float g  = -s + 2.0f * alpha * xv * pt * (1.0f / NDIM);
            const float tl = xv - STEP_PRI * g;
            float nx = fabsf(tl) - REG_SP * STEP_PRI;      // soft-threshold
            nx = nx > 0.0f ? nx : 0.0f;
            float om = 1.0f - nx;                          // clamp <= 1
            om = om > 0.0f ? om : 0.0f;
            xreg[k] = 1.0f - om;
        }
        __syncthreads();   // done reading b1/s_alpha before they are rewritten

        // ---- h = tr((I + x*x/N)^30)/N - 1 ; alpha += STEP_DUAL * h ----
#pragma unroll
        for (int k = 0; k < PER_THR; ++k) {
            const int r = 4 * k + r0;
            const float v = xreg[k];
            b0[r * STRIDE + c0] = v * v * (1.0f / NDIM) + (r == c0 ? 1.0f : 0.0f);
        }
        matmul64(b1, b0, b0, lane, wave);  // m^2
        matmul64(b2, b1, b1, lane, wave);  // m^4
        matmul64(b0, b1, b2, lane, wave);  // R = m^6      (m dead)
        matmul64(b1, b2, b2, lane, wave);  // m^8
        matmul64(b2, b0, b1, lane, wave);  // R = m^14
        matmul64(b0, b1, b1, lane, wave);  // m^16
        matmul64(b1, b2, b0, lane, wave);  // Q = m^30 -> b1
        if (tid == 0) {
            float tr = 0.0f;
            for (int i = 0; i < NDIM; ++i) tr += b1[i * STRIDE + i];
            s_alpha += STEP_DUAL * (tr * (1.0f / NDIM) - 1.0f);
        }
        __syncthreads();
    }

    // ---- out = threshold(x) ----
#pragma unroll
    for (int k = 0; k < PER_THR; ++k) {
        const float xv = xreg[k];
        outb[k * NTHREADS + tid] = (xv > 0.5f) ? xv : 0.0f;
    }
}

extern "C" void kernel_launch(void* const* d_in, const int* in_sizes, int n_in,
                              void* d_out, int out_size, void* d_ws, size_t ws_size,
                              hipStream_t stream) {
    (void)n_in; (void)d_ws; (void)ws_size; (void)out_size;
    const float* adj = (const float*)d_in[0];
    float* out = (float*)d_out;
    const int nb = in_sizes[0] / NELEM;   // 512 batch matrices, 1 workgroup each
    dag_prox_kernel<<<dim3(nb), dim3(NTHREADS), 0, stream>>>(adj, out);
}